// DecorrelatedReNorm_17231408791729
// MI455X (gfx1250) — compile-verified
//
#include <hip/hip_runtime.h>

typedef __attribute__((ext_vector_type(2))) float v2f;
typedef __attribute__((ext_vector_type(8))) float v8f;

#define C_DIM 512
#define N_ROWS 131072
#define BM 128
#define BN 64
#define BK 16
#define NK (C_DIM / BK)
#define LDA 20    // A tile LDS row stride (floats): 16B-aligned quads, conflict-free frags
#define LDB 18    // B tile LDS row stride (floats), [col][k] layout: contiguous v2f frags
#define NBUF 3    // pipeline depth (chunks in flight); (NK-2) % NBUF == 0

// bias[j] = sum_k running_mean[k] * running_W[k][j]   (512x512 matvec, tiny)
__global__ void zca_bias_kernel(const float* __restrict__ rm,
                                const float* __restrict__ W,
                                float* __restrict__ bias) {
    int j = blockIdx.x * blockDim.x + threadIdx.x;   // 0..511
    float acc = 0.f;
    #pragma unroll 8
    for (int k = 0; k < C_DIM; ++k) acc += rm[k] * W[k * C_DIM + j];
    bias[j] = acc;
}

// 16B global -> LDS copy through the async path (no VGPR staging, ASYNCcnt-tracked)
__device__ __forceinline__ void async_cp16(unsigned int lds_off, const float* g) {
    asm volatile("global_load_async_to_lds_b128 %0, %1, off"
                 :: "v"(lds_off), "v"((unsigned long long)(uintptr_t)g)
                 : "memory");
}

// one K-chunk of WMMA work: 4 k-steps x 2x2 tiles of V_WMMA_F32_16X16X4_F32
__device__ __forceinline__ void compute_chunk(const float* __restrict__ pA,
                                              const float* __restrict__ pB,
                                              int wm, int lhalf, int l16,
                                              int cB0, int cB1, v8f (&acc)[2][2]) {
    #pragma unroll
    for (int ks = 0; ks < 4; ++ks) {
        // f32 A/B layout: lanes 0-15 hold K={0,1}, lanes 16-31 hold K={2,3}
        const int kk = ks * 4 + lhalf * 2;
        v2f a0 = *(const v2f*)(pA + (wm * 32 +      l16) * LDA + kk);
        v2f a1 = *(const v2f*)(pA + (wm * 32 + 16 + l16) * LDA + kk);
        v2f b0 = *(const v2f*)(pB + cB0 * LDB + kk);
        v2f b1 = *(const v2f*)(pB + cB1 * LDB + kk);
        acc[0][0] = __builtin_amdgcn_wmma_f32_16x16x4_f32(false, a0, false, b0, (short)0, acc[0][0], false, false);
        acc[0][1] = __builtin_amdgcn_wmma_f32_16x16x4_f32(false, a0, false, b1, (short)0, acc[0][1], false, false);
        acc[1][0] = __builtin_amdgcn_wmma_f32_16x16x4_f32(false, a1, false, b0, (short)0, acc[1][0], false, false);
        acc[1][1] = __builtin_amdgcn_wmma_f32_16x16x4_f32(false, a1, false, b1, (short)0, acc[1][1], false, false);
    }
}

// out[n][j] = sum_k X[n][k] * W[k][j] - bias[j]
__launch_bounds__(256, 2)
__global__ void zca_gemm_kernel(const float* __restrict__ X,
                                const float* __restrict__ W,
                                const float* __restrict__ bias,
                                float* __restrict__ out) {
    __shared__ __align__(16) float sA[NBUF][BM * LDA];   // [row][k] per chunk of 16
    __shared__ __align__(16) float sB[NBUF][BN * LDB];   // [col][k] (K-transposed)

    const int tid   = threadIdx.x;      // 0..255
    const int lane  = tid & 31;
    const int wave  = tid >> 5;         // 0..7
    const int wm    = wave & 3;         // 32-row strip within 128
    const int wn    = wave >> 2;        // 32-col strip within 64
    const int lhalf = (lane < 16) ? 0 : 1;
    const int l16   = lane & 15;

    const long row0 = (long)blockIdx.x * BM;
    const int  col0 = blockIdx.y * BN;

    // accumulators pre-loaded with -bias (epilogue becomes a pure store)
    const float nb0 = -bias[col0 + wn * 32 + l16];
    const float nb1 = -bias[col0 + wn * 32 + 16 + l16];
    v8f acc[2][2];
    #pragma unroll
    for (int i = 0; i < 8; ++i) {
        acc[0][0][i] = nb0; acc[1][0][i] = nb0;
        acc[0][1][i] = nb1; acc[1][1][i] = nb1;
    }

    // A: 128x16 floats = 512 x 16B -> 2 async copies per thread
    const int arow0 = tid >> 2;          // 0..63
    const int aq    = (tid & 3) * 4;     // 0,4,8,12
    const int arow1 = arow0 + 64;
    // B: 16x64 floats -> 4 scalar loads per thread (coalesced over cols), transposed store
    const int bc  = tid & 63;            // col within tile
    const int bk4 = tid >> 6;            // 0..3 (k sub-row)

    const float* gA0 = X + (row0 + arow0) * C_DIM + aq;
    const float* gA1 = X + (row0 + arow1) * C_DIM + aq;

    // LDS byte offsets (low 32 bits of the generic shared pointer = LDS offset)
    const unsigned int a0off = (unsigned int)(uintptr_t)&sA[0][arow0 * LDA + aq];
    const unsigned int a1off = (unsigned int)(uintptr_t)&sA[0][arow1 * LDA + aq];
    const unsigned int aStride = BM * LDA * 4;   // bytes per A buffer

    // prologue: A chunks 0,1 in flight; B chunks 0,1 staged through registers
    async_cp16(a0off + 0 * aStride, gA0);
    async_cp16(a1off + 0 * aStride, gA1);
    async_cp16(a0off + 1 * aStride, gA0 + BK);
    async_cp16(a1off + 1 * aStride, gA1 + BK);
    #pragma unroll
    for (int i = 0; i < 4; ++i) {
        const int k = bk4 + i * 4;
        sB[0][bc * LDB + k] = W[(size_t)k * C_DIM + col0 + bc];
        sB[1][bc * LDB + k] = W[(size_t)(BK + k) * C_DIM + col0 + bc];
    }
    asm volatile("s_wait_asynccnt 0x2" ::: "memory");   // A chunk 0 landed
    __syncthreads();

    const int cB0 = wn * 32 + l16;       // B fragment columns (within tile)
    const int cB1 = cB0 + 16;

    // running per-thread pointers at chunk kc+2 (the chunk each stage issues)
    const float* gA0i = gA0 + 2 * BK;
    const float* gA1i = gA1 + 2 * BK;
    const float* gBi  = W + (size_t)(2 * BK + bk4) * C_DIM + col0 + bc;

    // One pipeline stage: compute buffer CB (= kc%3) and issue buffer IB (= (kc+2)%3)
    // are compile-time constants; J is the chunk offset within the macro-iteration.
#define STAGE(CB, IB, J)  do {                                                    \
        async_cp16(a0off + (IB) * aStride, gA0i + (J) * BK);                      \
        async_cp16(a1off + (IB) * aStride, gA1i + (J) * BK);                      \
        float pb[4];                                                              \
        _Pragma("unroll")                                                         \
        for (int i = 0; i < 4; ++i)                                               \
            pb[i] = gBi[(size_t)((J) * BK + i * 4) * C_DIM];                      \
        __builtin_prefetch(gA0i + ((J) + 2) * BK, 0, 1);                          \
        compute_chunk(sA[CB], sB[CB], wm, lhalf, l16, cB0, cB1, acc);             \
        {                                                                         \
            float* __restrict__ dB = sB[IB];                                      \
            _Pragma("unroll")                                                     \
            for (int i = 0; i < 4; ++i) dB[bc * LDB + bk4 + i * 4] = pb[i];       \
        }                                                                         \
        asm volatile("s_wait_asynccnt 0x2" ::: "memory");                         \
        __syncthreads();                                                          \
    } while (0)

    // steady state: kc = 0..NK-3 in groups of 3 -> all buffer indices constant
    for (int kt = 0; kt < (NK - 2) / 3; ++kt) {
        STAGE(0, 2, 0);
        STAGE(1, 0, 1);
        STAGE(2, 1, 2);
        gA0i += 3 * BK;
        gA1i += 3 * BK;
        gBi  += (size_t)3 * BK * C_DIM;
    }
#undef STAGE

    // drain: chunk NK-2 (buf 0), then chunk NK-1 (buf 1)
    compute_chunk(sA[0], sB[0], wm, lhalf, l16, cB0, cB1, acc);
    asm volatile("s_wait_asynccnt 0x0" ::: "memory");   // last chunk landed
    __syncthreads();
    compute_chunk(sA[1], sB[1], wm, lhalf, l16, cB0, cB1, acc);

    // epilogue: 16x16 f32 C/D layout -> VGPR r holds row r (lanes 0-15) / r+8 (lanes 16-31)
    #pragma unroll
    for (int mi = 0; mi < 2; ++mi) {
        const long rbase = row0 + wm * 32 + mi * 16 + lhalf * 8;
        #pragma unroll
        for (int ni = 0; ni < 2; ++ni) {
            const int c = col0 + wn * 32 + ni * 16 + l16;
            #pragma unroll
            for (int r = 0; r < 8; ++r) {
                out[(rbase + r) * C_DIM + c] = acc[mi][ni][r];
            }
        }
    }
}

extern "C" void kernel_launch(void* const* d_in, const int* in_sizes, int n_in,
                              void* d_out, int out_size, void* d_ws, size_t ws_size,
                              hipStream_t stream) {
    const float* X  = (const float*)d_in[0];   // [131072, 512]
    const float* rm = (const float*)d_in[1];   // [512]
    const float* W  = (const float*)d_in[2];   // [512, 512]
    float* out  = (float*)d_out;               // [131072, 512]
    float* bias = (float*)d_ws;                // 512 floats of scratch

    zca_bias_kernel<<<dim3(C_DIM / 256), dim3(256), 0, stream>>>(rm, W, bias);

    dim3 grid(N_ROWS / BM, C_DIM / BN);        // (1024, 8)
    zca_gemm_kernel<<<grid, dim3(256), 0, stream>>>(X, W, bias, out);
}